// ImageCaptioningModel_12970801233935
// MI455X (gfx1250) — compile-verified
//
#include <hip/hip_runtime.h>
#include <math.h>

typedef float v2f __attribute__((ext_vector_type(2)));
typedef float v8f __attribute__((ext_vector_type(8)));

#define B_   64
#define T_   32
#define V_   32000
#define E_   128
#define H_   256
#define F_   512
#define G4   (4 * H_)    // 1024
#define KIN  (E_ + F_)   // 640
#define MPRE (B_ * T_)   // 2048

__device__ __forceinline__ v8f wmma_f32(v2f a, v2f b, v8f c) {
    return __builtin_amdgcn_wmma_f32_16x16x4_f32(false, a, false, b, (short)0, c, false, false);
}

__device__ __forceinline__ float sigmoidf_(float x) { return 1.0f / (1.0f + expf(-x)); }

// ---------------------------------------------------------------------------
// Kernel A: pre[t*B + b, j] = [emb[cap[b,t]], features[b]] . W_ih[j,:] + b_ih[j] + b_hh[j]
// GEMM (2048 x 640) x (640 x 1024), one 16x16 tile per wave32.
// ---------------------------------------------------------------------------
__global__ __launch_bounds__(128) void pre_gates_kernel(
    const float* __restrict__ features, const int* __restrict__ captions,
    const float* __restrict__ emb, const float* __restrict__ W_ih,
    const float* __restrict__ b_ih, const float* __restrict__ b_hh,
    float* __restrict__ pre)
{
    const int wave = threadIdx.x >> 5;
    const int lane = threadIdx.x & 31;
    const int tile = blockIdx.x * 4 + wave;          // 128 m-tiles x 64 n-tiles
    const int tm = tile >> 6;
    const int tn = tile & 63;
    const int lidx = lane & 15;
    const int lhalf = lane >> 4;

    const int m = tm * 16 + lidx;                    // row = t*B + b
    const int t = m >> 6;
    const int b = m & 63;
    const float* __restrict__ arow_e = emb + (size_t)captions[b * T_ + t] * E_;
    const float* __restrict__ arow_f = features + (size_t)b * F_;
    const int ncol = tn * 16 + lidx;
    const float* __restrict__ bcol = W_ih + (size_t)ncol * KIN;  // B[k][n] = W_ih[n*KIN + k]

    v8f acc = {};
    #pragma unroll 8
    for (int k0 = 0; k0 < E_; k0 += 4) {             // embedding slice of xin
        const int k = k0 + lhalf * 2;
        v2f a;  a.x = arow_e[k];     a.y = arow_e[k + 1];
        v2f bb; bb.x = bcol[k];      bb.y = bcol[k + 1];
        acc = wmma_f32(a, bb, acc);
    }
    #pragma unroll 8
    for (int k0 = 0; k0 < F_; k0 += 4) {             // context slice (== features, exactly)
        const int k = k0 + lhalf * 2;
        v2f a;  a.x = arow_f[k];        a.y = arow_f[k + 1];
        v2f bb; bb.x = bcol[E_ + k];    bb.y = bcol[E_ + k + 1];
        acc = wmma_f32(a, bb, acc);
    }

    const float bias = b_ih[ncol] + b_hh[ncol];
    #pragma unroll
    for (int v = 0; v < 8; ++v) {
        const int row = tm * 16 + lhalf * 8 + v;
        pre[(size_t)row * G4 + ncol] = acc[v] + bias;
    }
}

// ---------------------------------------------------------------------------
// Kernel B1 (per step t): gates[b, j] = pre[t*B + b, j] + h[b,:] . W_hh[j,:]
// GEMM (64 x 256) x (256 x 1024), 256 tiles, one per wave.
// ---------------------------------------------------------------------------
__global__ __launch_bounds__(128) void gates_kernel(
    const float* __restrict__ hbuf, const float* __restrict__ W_hh,
    const float* __restrict__ pre, float* __restrict__ gates, int t)
{
    const int wave = threadIdx.x >> 5;
    const int lane = threadIdx.x & 31;
    const int tile = blockIdx.x * 4 + wave;          // 4 m-tiles x 64 n-tiles
    const int tm = tile >> 6;
    const int tn = tile & 63;
    const int lidx = lane & 15;
    const int lhalf = lane >> 4;

    const float* __restrict__ arow = hbuf + (size_t)(tm * 16 + lidx) * H_;
    const int ncol = tn * 16 + lidx;
    const float* __restrict__ bcol = W_hh + (size_t)ncol * H_;

    v8f acc = {};
    #pragma unroll 8
    for (int k0 = 0; k0 < H_; k0 += 4) {
        const int k = k0 + lhalf * 2;
        v2f a;  a.x = arow[k];  a.y = arow[k + 1];
        v2f bb; bb.x = bcol[k]; bb.y = bcol[k + 1];
        acc = wmma_f32(a, bb, acc);
    }

    const float* __restrict__ prow = pre + (size_t)t * B_ * G4;
    #pragma unroll
    for (int v = 0; v < 8; ++v) {
        const int row = tm * 16 + lhalf * 8 + v;
        gates[(size_t)row * G4 + ncol] = acc[v] + prow[(size_t)row * G4 + ncol];
    }
}

// ---------------------------------------------------------------------------
// Kernel B2 (per step t): LSTM cell elementwise, PyTorch gate order i,f,g,o.
// Writes h into running buffer and into h_all[(b*T + t), :].
// ---------------------------------------------------------------------------
__global__ __launch_bounds__(256) void lstm_cell_kernel(
    const float* __restrict__ gates, float* __restrict__ hbuf,
    float* __restrict__ cbuf, float* __restrict__ hall, int t)
{
    const int idx = blockIdx.x * 256 + threadIdx.x;  // 0 .. B_*H_-1
    const int b = idx >> 8;
    const int j = idx & 255;
    const float* __restrict__ g = gates + (size_t)b * G4;

    const float ig = sigmoidf_(g[j]);
    const float fg = sigmoidf_(g[H_ + j]);
    const float gg = tanhf(g[2 * H_ + j]);
    const float og = sigmoidf_(g[3 * H_ + j]);

    const float c = fg * cbuf[idx] + ig * gg;
    cbuf[idx] = c;
    const float h = og * tanhf(c);
    hbuf[idx] = h;
    hall[((size_t)b * T_ + t) * H_ + j] = h;
}

// ---------------------------------------------------------------------------
// Kernel C: out[(b*T + t), v] = h_all[(b*T + t), :] . fc_W[v, :] + fc_b[v]
// GEMM (2048 x 256) x (256 x 32000). Register blocking: each wave computes
// 2 M-tiles x 4 N-tiles (32 rows x 64 cols). Each B fragment feeds 2 WMMAs
// (halves fc_W L2 traffic vs 1x4), each A fragment feeds 4 WMMAs.
// ---------------------------------------------------------------------------
__global__ __launch_bounds__(256) void logits_kernel(
    const float* __restrict__ hall, const float* __restrict__ fc_W,
    const float* __restrict__ fc_b, float* __restrict__ out)
{
    const int wave = threadIdx.x >> 5;
    const int lane = threadIdx.x & 31;
    const int wid = blockIdx.x * 8 + wave;           // 64 m-blocks x 500 n-strips
    const int tmb = wid / 500;                       // 0..63  (32 rows each)
    const int ts  = wid % 500;                       // 0..499 (64 cols each)
    const int lidx = lane & 15;
    const int lhalf = lane >> 4;

    const float* __restrict__ arow0 = hall + (size_t)(tmb * 32 + lidx) * H_;
    const float* __restrict__ arow1 = arow0 + 16 * H_;
    const int n0 = ts * 64;
    const float* __restrict__ bc0 = fc_W + (size_t)(n0 + lidx) * H_;
    const float* __restrict__ bc1 = bc0 + 16 * H_;
    const float* __restrict__ bc2 = bc0 + 32 * H_;
    const float* __restrict__ bc3 = bc0 + 48 * H_;

    v8f a00 = {}, a01 = {}, a02 = {}, a03 = {};      // M-tile 0 x N-tiles 0..3
    v8f a10 = {}, a11 = {}, a12 = {}, a13 = {};      // M-tile 1 x N-tiles 0..3
    #pragma unroll 4
    for (int k0 = 0; k0 < H_; k0 += 4) {
        const int k = k0 + lhalf * 2;
        v2f a0; a0.x = arow0[k]; a0.y = arow0[k + 1];
        v2f a1; a1.x = arow1[k]; a1.y = arow1[k + 1];
        v2f b0; b0.x = bc0[k]; b0.y = bc0[k + 1];
        v2f b1; b1.x = bc1[k]; b1.y = bc1[k + 1];
        v2f b2; b2.x = bc2[k]; b2.y = bc2[k + 1];
        v2f b3; b3.x = bc3[k]; b3.y = bc3[k + 1];
        a00 = wmma_f32(a0, b0, a00);
        a10 = wmma_f32(a1, b0, a10);
        a01 = wmma_f32(a0, b1, a01);
        a11 = wmma_f32(a1, b1, a11);
        a02 = wmma_f32(a0, b2, a02);
        a12 = wmma_f32(a1, b2, a12);
        a03 = wmma_f32(a0, b3, a03);
        a13 = wmma_f32(a1, b3, a13);
    }

    const float fb0 = fc_b[n0 + lidx];
    const float fb1 = fc_b[n0 + 16 + lidx];
    const float fb2 = fc_b[n0 + 32 + lidx];
    const float fb3 = fc_b[n0 + 48 + lidx];
    #pragma unroll
    for (int v = 0; v < 8; ++v) {
        const size_t r0 = (size_t)(tmb * 32 + lhalf * 8 + v);
        float* __restrict__ orow0 = out + r0 * V_;
        float* __restrict__ orow1 = orow0 + (size_t)16 * V_;
        orow0[n0 + lidx]      = a00[v] + fb0;
        orow0[n0 + 16 + lidx] = a01[v] + fb1;
        orow0[n0 + 32 + lidx] = a02[v] + fb2;
        orow0[n0 + 48 + lidx] = a03[v] + fb3;
        orow1[n0 + lidx]      = a10[v] + fb0;
        orow1[n0 + 16 + lidx] = a11[v] + fb1;
        orow1[n0 + 32 + lidx] = a12[v] + fb2;
        orow1[n0 + 48 + lidx] = a13[v] + fb3;
    }
}

// ---------------------------------------------------------------------------
extern "C" void kernel_launch(void* const* d_in, const int* in_sizes, int n_in,
                              void* d_out, int out_size, void* d_ws, size_t ws_size,
                              hipStream_t stream) {
    const float* features = (const float*)d_in[0];
    const int*   captions = (const int*)d_in[1];
    const float* emb      = (const float*)d_in[2];
    // d_in[3..6] = attn_W / attn_b / attn_v / attn_vb: unused.
    // enc is features broadcast over T, so all T attention scores are equal,
    // softmax is exactly uniform (1/32, a power of two) and context == features.
    const float* W_ih = (const float*)d_in[7];
    const float* W_hh = (const float*)d_in[8];
    const float* b_ih = (const float*)d_in[9];
    const float* b_hh = (const float*)d_in[10];
    const float* fc_W = (const float*)d_in[11];
    const float* fc_b = (const float*)d_in[12];
    float* out = (float*)d_out;

    char* ws = (char*)d_ws;
    float* pre   = (float*)ws;                                   // 2048*1024 f32 = 8 MB
    float* hall  = (float*)(ws + (size_t)MPRE * G4 * 4);         // 2048*256 f32 = 2 MB
    float* hbuf  = hall + (size_t)MPRE * H_;                     // 64*256
    float* cbuf  = hbuf + B_ * H_;                               // 64*256
    float* gates = cbuf + B_ * H_;                               // 64*1024

    // h0 = c0 = 0 (contiguous h/c buffers, one capturable memset)
    hipMemsetAsync(hbuf, 0, (size_t)2 * B_ * H_ * sizeof(float), stream);

    // All-timesteps input-side gates: (2048x640)x(640x1024)
    pre_gates_kernel<<<2048, 128, 0, stream>>>(features, captions, emb, W_ih,
                                               b_ih, b_hh, pre);

    // Sequential recurrence: tiny hidden GEMM + cell per step
    for (int t = 0; t < T_; ++t) {
        gates_kernel<<<64, 128, 0, stream>>>(hbuf, W_hh, pre, gates, t);
        lstm_cell_kernel<<<64, 256, 0, stream>>>(gates, hbuf, cbuf, hall, t);
    }

    // One big vocab projection: (2048x256)x(256x32000) -> d_out [B,T,V]
    logits_kernel<<<4000, 256, 0, stream>>>(hall, fc_W, fc_b, out);
}